// SSEMultiPartitionState_89300960019113
// MI455X (gfx1250) — compile-verified
//
#include <hip/hip_runtime.h>

typedef __attribute__((ext_vector_type(2))) float v2f;
typedef __attribute__((ext_vector_type(8))) float v8f;

#define BB 4
#define SS 2048
#define KK 2
#define CC 64
#define DD 128
#define PP 1024

// -----------------------------------------------------------------------------
// Kernel 1: acc[p,d] = (1/C) * sum_c states[p,c,d]
// Uses V_WMMA_F32_16X16X4_F32 with an all-ones A matrix so every row of the
// 16x16 accumulator is the column-sum of the 4x16 B chunk; accumulating over
// 16 chunks (c = 0..63 step 4) yields the full reduction over C.
// -----------------------------------------------------------------------------
__global__ __launch_bounds__(256) void k_states_mean(const float* __restrict__ states,
                                                     float* __restrict__ acc) {
    __shared__ float tile[CC * DD]; // 64 x 128 f32 = 32 KB
    const int p = blockIdx.x;
    const int t = threadIdx.x;

    // Cooperative fully-coalesced 128-bit load of the tile into LDS.
    const float4* src = (const float4*)(states + (size_t)p * (CC * DD));
    float4* dst = (float4*)tile;
#pragma unroll
    for (int j = 0; j < (CC * DD / 4) / 256; ++j)
        dst[t + j * 256] = src[t + j * 256];
    __syncthreads();

    const int lane  = t & 31;
    const int wave  = t >> 5;     // 8 waves -> 8 d-tiles of 16
    const int n     = lane & 15;  // column within the d tile
    const int kh    = lane >> 4;  // which K-pair this lane supplies
    const int dbase = wave * 16;

    v2f a; a[0] = 1.0f; a[1] = 1.0f;   // all-ones 16x4 A matrix
    v8f c = {};
#pragma unroll
    for (int cc = 0; cc < CC; cc += 4) {
        const int row = cc + (kh << 1);
        v2f b;
        b[0] = tile[row * DD + dbase + n];
        b[1] = tile[(row + 1) * DD + dbase + n];
        c = __builtin_amdgcn_wmma_f32_16x16x4_f32(false, a, false, b,
                                                  (short)0, c, false, false);
    }

    // D layout: VGPR0, lanes 0-15 hold M=0, N=lane. All rows are identical
    // (A was all ones), so lane n of c[0] = sum_c states[p,c,dbase+n].
    if (lane < 16)
        acc[p * DD + dbase + lane] = c[0] * (1.0f / CC);
}

// -----------------------------------------------------------------------------
// Kernel 2: scatter. For each token (b,s): ksum_k = sum_c keys[b,s,k,c];
// acc[idx[b,s,k], d] += (ksum_k / C) * values[b,s,d]
// -----------------------------------------------------------------------------
__device__ __forceinline__ void atomic_add_f32(float* p, float v) {
    unsafeAtomicAdd(p, v);   // lowers to global_atomic_add_f32 on gfx1250
}

__global__ __launch_bounds__(128) void k_scatter(const int* __restrict__ idx,
                                                 const float* __restrict__ keys,
                                                 const float* __restrict__ values,
                                                 float* __restrict__ acc) {
    __shared__ float red[KK * CC]; // 128 floats: [k*64 + c]
    const int bs = blockIdx.x;     // 0 .. B*S-1
    const int t  = threadIdx.x;    // 0 .. 127

    red[t] = keys[bs * (KK * CC) + t];
    __syncthreads();
    // Tree reduction within each 64-wide segment.
#pragma unroll
    for (int off = 32; off > 0; off >>= 1) {
        if ((t & 63) < off) red[t] += red[t + off];
        __syncthreads();
    }
    const float ks0 = red[0]  * (1.0f / CC);
    const float ks1 = red[CC] * (1.0f / CC);

    const int i0 = idx[bs * KK + 0];
    const int i1 = idx[bs * KK + 1];
    const float v = values[bs * DD + t];   // t == d here (DD == 128 threads)

    atomic_add_f32(&acc[i0 * DD + t], ks0 * v);
    atomic_add_f32(&acc[i1 * DD + t], ks1 * v);
}

// -----------------------------------------------------------------------------
// Kernel 3: gather + weight.
// out[b,s,d] = queries[b,s,d] * (acc[idx[b,s,0],d] + acc[idx[b,s,1],d])
// -----------------------------------------------------------------------------
__global__ __launch_bounds__(256) void k_gather(const int* __restrict__ idx,
                                                const float* __restrict__ queries,
                                                const float* __restrict__ acc,
                                                float* __restrict__ out) {
    const int gid = blockIdx.x * 256 + threadIdx.x; // 0 .. B*S*D-1
    const int bs  = gid >> 7;
    const int d   = gid & (DD - 1);
    const int i0  = idx[bs * KK + 0];
    const int i1  = idx[bs * KK + 1];
    out[gid] = queries[gid] * (acc[i0 * DD + d] + acc[i1 * DD + d]);
}

// -----------------------------------------------------------------------------
extern "C" void kernel_launch(void* const* d_in, const int* in_sizes, int n_in,
                              void* d_out, int out_size, void* d_ws, size_t ws_size,
                              hipStream_t stream) {
    const int*   idx     = (const int*)d_in[0];   // (B,S,K) int32
    const float* keys    = (const float*)d_in[1]; // (B,S,K,C) f32
    const float* values  = (const float*)d_in[2]; // (B,S,D) f32
    const float* queries = (const float*)d_in[3]; // (B,S,D) f32
    const float* states  = (const float*)d_in[4]; // (P,C,D) f32
    float*       out     = (float*)d_out;         // (B,S,D) f32
    float*       acc     = (float*)d_ws;          // (P,D) f32 = 512 KB scratch

    k_states_mean<<<PP, 256, 0, stream>>>(states, acc);
    k_scatter<<<BB * SS, 128, 0, stream>>>(idx, keys, values, acc);
    k_gather<<<(BB * SS * DD) / 256, 256, 0, stream>>>(idx, queries, acc, out);
}